// GetCostVolume_1915555414484
// MI455X (gfx1250) — compile-verified
//
#include <hip/hip_runtime.h>

// Cost-volume forward for MI455X (gfx1250, wave32).
//
//   out[b, p, d, h, w] = (w >= d) * (1/k_p^2) * sum_{c < k_p^2} |x[b,c,h,w] - y[b,c,h,w-d]|
//   k_p = p+3, p = 0..8;   B=2, C=121, H=128, W=256, D=48.
//
// Roofline: 113 MB out + 64 MB in ≈ 177 MB minimum HBM traffic (~7.6 us @ 23.3 TB/s);
// only ~12.5 GFLOP of f32 MACs -> memory bound, so stay in exact f32.
//
// Strategy: one workgroup per (b,h). Stage the full (C,W) rows of x and y into
// 320KB WGP LDS, transposed to [w][c] (row stride 132 floats: 8B-aligned channel
// pairs, bank step 132 % 64 = 4 -> 64 distinct banks per wave-wide b64 op).
// Staging uses CDNA5 GLOBAL_LOAD_ASYNC_TO_LDS_B32: per-lane global address +
// per-lane LDS address, so the transpose happens in the async unit with no VGPR
// round-trip (ASYNCcnt-tracked, drained with s_wait_asynccnt before the barrier).
// The 9 prefix-means are a constant 9x121 matrix times the per-pixel diff
// vector -> exact-precision V_WMMA_F32_16X16X4_F32, K over 31 chunks of 4.

typedef __attribute__((ext_vector_type(2))) float v2f;
typedef __attribute__((ext_vector_type(8))) float v8f;

#define B_   2
#define C_   121
#define H_   128
#define W_   256
#define D_   48
#define P_   9
#define CPAD 124      // channels padded to 31 * 4 (zero-filled pad)
#define CST  132      // LDS [w][c] row stride in floats

__global__ __launch_bounds__(256)
void cost_volume_wmma_kernel(const float* __restrict__ x,
                             const float* __restrict__ y,
                             float* __restrict__ out) {
  extern __shared__ float lds[];            // 2 * 256 * 132 * 4B = 270,336 B
  float* ldsx = lds;                        // [W_][CST]
  float* ldsy = lds + W_ * CST;

  const int bh  = blockIdx.x;               // 0 .. B*H-1
  const int b   = bh / H_;
  const int h   = bh - b * H_;
  const int tid = threadIdx.x;

  // ---- Stage x[b,:,h,:], y[b,:,h,:] into LDS transposed to [w][c] ----
  // Async global->LDS scatter: global side coalesced over w, LDS side strided
  // by CST. Each HBM input byte is read exactly once across the whole kernel.
  const size_t rowbase = ((size_t)b * C_ * H_ + (size_t)h) * W_;   // + c*H*W + w
  const unsigned gss = __builtin_amdgcn_groupstaticsize();         // dyn-LDS base
  for (int idx = tid; idx < C_ * W_; idx += 256) {    // 121*256 / 256 = 121 iters
    const int c = idx >> 8;                 // idx / W_
    const int w = idx & (W_ - 1);
    const unsigned long long gx =
        (unsigned long long)(x + rowbase + (size_t)c * (H_ * W_) + (size_t)w);
    const unsigned long long gy =
        (unsigned long long)(y + rowbase + (size_t)c * (H_ * W_) + (size_t)w);
    const unsigned lx = gss + (unsigned)(w * CST + c) * 4u;        // ldsx offset
    const unsigned ly = lx + (unsigned)(W_ * CST) * 4u;            // ldsy offset
    asm volatile("global_load_async_to_lds_b32 %0, %1, off"
                 :: "v"(lx), "v"(gx) : "memory");
    asm volatile("global_load_async_to_lds_b32 %0, %1, off"
                 :: "v"(ly), "v"(gy) : "memory");
  }
  // Zero-fill pad channels 121..123 (garbage NaN * 0-weight would poison WMMA).
  for (int idx = tid; idx < (CPAD - C_) * W_; idx += 256) {        // 3 iters
    const int c = C_ + (idx >> 8);
    const int w = idx & (W_ - 1);
    ldsx[w * CST + c] = 0.0f;
    ldsy[w * CST + c] = 0.0f;
  }
  // Drain this wave's async copies, then publish LDS to the workgroup.
  asm volatile("s_wait_asynccnt 0" ::: "memory");
  __syncthreads();

  // ---- Per-lane constants (wave32 WMMA operand layout, ISA 7.12.2) ----
  const int wave = tid >> 5;
  const int lane = tid & 31;
  const int half = lane >> 4;               // 0: K={0,1}, 1: K={2,3}
  const int l16  = lane & 15;               // A row m / B,D column n-offset
  const int kofs = half << 1;

  // Weight row m = l16 of the constant 9x121 prefix-mean matrix.
  const int   kk   = l16 + 3;
  const int   ksq  = (l16 < P_) ? kk * kk : 0;              // 0 -> all-zero row
  const float inv  = (l16 < P_) ? 1.0f / (float)(kk * kk) : 0.0f;

  const size_t pStride = (size_t)D_ * H_ * W_;
  const size_t dStride = (size_t)H_ * W_;
  const size_t outBase = (size_t)b * P_ * pStride + (size_t)h * W_;

  // ---- 768 (d, w-tile) tiles split over the 8 waves ----
  for (int t = wave; t < D_ * (W_ / 16); t += 8) {
    const int d     = t >> 4;
    const int wbase = (t & 15) << 4;
    const int n     = wbase + l16;          // this lane's output column (w)
    const int wy    = n - d;
    const float* xp = ldsx + n * CST + kofs;
    const float* yp = ldsy + (wy < 0 ? 0 : wy) * CST + kofs;  // clamped; masked below

    v8f acc = {};
    for (int chunk = 0; chunk < CPAD / 4; ++chunk) {
      const int c0 = (chunk << 2) + kofs;   // this lane's channel pair base
      // A (16x4 weights): lane holds (m, K=kofs..kofs+1)
      v2f a;
      a.x = (c0     < ksq) ? inv : 0.0f;
      a.y = (c0 + 1 < ksq) ? inv : 0.0f;
      // B (4x16 diffs): one b64 from each LDS image, |x - y_shift|
      const v2f xv = *(const v2f*)(xp + (chunk << 2));
      const v2f yv = *(const v2f*)(yp + (chunk << 2));
      v2f bm;
      bm.x = __builtin_fabsf(xv.x - yv.x);
      bm.y = __builtin_fabsf(xv.y - yv.y);
      // D = A x B + C   (f32, exact-precision path)
      acc = __builtin_amdgcn_wmma_f32_16x16x4_f32(
          /*neg_a=*/false, a, /*neg_b=*/false, bm,
          /*c_mod=*/(short)0, acc, /*reuse_a=*/false, /*reuse_b=*/false);
    }

    // D layout: VGPR r, lanes 0-15 -> row r, lanes 16-31 -> row r+8.
    // Valid prefixes p = 0..8; zero the w < d strip; stream out (never re-read).
    const bool on = (n >= d);
    const size_t ob = outBase + (size_t)d * dStride + (size_t)n;
#pragma unroll
    for (int r = 0; r < 8; ++r) {
      const int p = r + (half << 3);
      if (p < P_) {
        __builtin_nontemporal_store(on ? acc[r] : 0.0f,
                                    out + ob + (size_t)p * pStride);
      }
    }
  }
}

extern "C" void kernel_launch(void* const* d_in, const int* in_sizes, int n_in,
                              void* d_out, int out_size, void* d_ws, size_t ws_size,
                              hipStream_t stream) {
  (void)in_sizes; (void)n_in; (void)out_size; (void)d_ws; (void)ws_size;
  const float* x = (const float*)d_in[0];
  const float* y = (const float*)d_in[1];
  float* out = (float*)d_out;

  const size_t shmem = (size_t)2 * W_ * CST * sizeof(float);   // 270,336 B < 320 KB
  dim3 grid(B_ * H_);                                          // 256 workgroups
  dim3 block(256);                                             // 8 wave32 waves
  cost_volume_wmma_kernel<<<grid, block, shmem, stream>>>(x, y, out);
}